// RichardsonDUN_34815004901606
// MI455X (gfx1250) — compile-verified
//
#include <hip/hip_runtime.h>

// CDNA5 (gfx1250) wave32 WMMA types
typedef __attribute__((ext_vector_type(2))) float v2f;
typedef __attribute__((ext_vector_type(8))) float v8f;

// D = A(16x4 f32) * B(4x16 f32) + C(16x16 f32)
#define WMMA_F32(A, B, C) \
  __builtin_amdgcn_wmma_f32_16x16x4_f32(false, (A), false, (B), (short)0, (C), false, false)

// Row stride 74 floats: B-frag half-wave groups differ by 148 mod 64 = 20 banks
// (disjoint windows), A-frag rows step 10 banks (distinct over 16 rows),
// update-store halves differ by 16 banks -> all conflict-free. Rows 8B-aligned.
#define LDS_STRIDE 74
#define MAT_FLOATS (64 * LDS_STRIDE)

__launch_bounds__(512)
__global__ void richardson_dun_kernel(const float* __restrict__ Agr_all,
                                      const float* __restrict__ Agi_all,
                                      const float* __restrict__ omegas,
                                      int nlayers,
                                      float* __restrict__ out)
{
    extern __shared__ float lds[];
    float* sAr = lds;                  // A real staging, 64 x 74
    float* sAi = lds + MAT_FLOATS;     // A imag staging
    float* sXr = lds + 2 * MAT_FLOATS; // X real (published each layer)
    float* sXi = lds + 3 * MAT_FLOATS; // X imag

    const int b      = blockIdx.x;
    const int tid    = threadIdx.x;          // 0..511 (16 wave32s)
    const int lane   = tid & 31;
    const int wave   = tid >> 5;             // 0..15
    const int ti     = wave >> 2;            // output row tile 0..3
    const int tj     = wave & 3;             // output col tile 0..3
    const int laneHi = lane >> 4;            // 0 for lanes 0-15, 1 for 16-31
    const int lane15 = lane & 15;

    const float* Agr = Agr_all + (size_t)b * 4096;
    const float* Agi = Agi_all + (size_t)b * 4096;

    // ---- Stage A into LDS (coalesced b64 global loads / b64 LDS stores) ----
    #pragma unroll
    for (int it = 0; it < 4; ++it) {
        int e = (tid + it * 512) * 2;        // even element index 0..4094
        int r = e >> 6, c = e & 63;          // c even -> 8B aligned in LDS
        *(v2f*)&sAr[r * LDS_STRIDE + c] = *(const v2f*)&Agr[e];
        *(v2f*)&sAi[r * LDS_STRIDE + c] = *(const v2f*)&Agi[e];
    }

    const int gm_base = 16 * ti + 8 * laneHi;            // + v gives output row
    const int gn      = 16 * tj + lane15;                // output col (fixed per lane)
    const int arowOff = (16 * ti + lane15) * LDS_STRIDE; // A-frag row
    const int acolOff = 2 * laneHi;                      // A-frag K pair base
    const int browOff = 2 * laneHi;                      // B-frag K pair base

    // ---- X tile lives in registers: layer 0 folded -> X = omega0 * I ----
    const float om0 = omegas[0];
    float xr[8], xi[8];
    #pragma unroll
    for (int v = 0; v < 8; ++v) {
        xr[v] = ((gm_base + v) == gn) ? om0 : 0.0f;
        xi[v] = 0.0f;
    }
    // Publish X to LDS (padding columns are never read, no zero-fill needed)
    #pragma unroll
    for (int v = 0; v < 8; ++v) {
        int xo = (gm_base + v) * LDS_STRIDE + gn;
        sXr[xo] = xr[v];
        sXi[xo] = xi[v];
    }
    __syncthreads();   // A staged + X published

    // ---- Layer-invariant A fragments in registers (Karatsuba: Ar, Ai, Ar+Ai) ----
    v2f ar[16], ai[16], as_[16];
    #pragma unroll
    for (int kk = 0; kk < 16; ++kk) {
        int off = arowOff + kk * 4 + acolOff;
        ar[kk]  = *(const v2f*)&sAr[off];
        ai[kk]  = *(const v2f*)&sAi[off];
        as_[kk] = ar[kk] + ai[kk];
    }

    // ---- Richardson layers 1..nlayers-1 ----
    // Complex matmul via 3 real matmuls (Karatsuba):
    //   P1 = Ar*Xr, P2 = Ai*Xi, P3 = (Ar+Ai)*(Xr+Xi)
    //   Cr = P1 - P2, Ci = P3 - P1 - P2
    for (int l = 1; l < nlayers; ++l) {
        const float omega = omegas[l];

        v8f p1 = {};
        v8f p2 = {};
        v8f p3 = {};

        #pragma unroll
        for (int kk = 0; kk < 16; ++kk) {
            // B fragments: lane holds X[c][N], X[c+1][N], c = 4*kk + 2*laneHi
            int bb = (4 * kk + browOff) * LDS_STRIDE + gn;
            v2f bxr; bxr.x = sXr[bb]; bxr.y = sXr[bb + LDS_STRIDE];
            v2f bxi; bxi.x = sXi[bb]; bxi.y = sXi[bb + LDS_STRIDE];
            v2f bxs = bxr + bxi;     // co-executes with the XDL pipe

            p1 = WMMA_F32(ar[kk],  bxr, p1);
            p2 = WMMA_F32(ai[kk],  bxi, p2);
            p3 = WMMA_F32(as_[kk], bxs, p3);
        }

        // Richardson update, entirely in registers
        #pragma unroll
        for (int v = 0; v < 8; ++v) {
            float cr  = p1[v] - p2[v];
            float ci  = p3[v] - p1[v] - p2[v];
            float idv = ((gm_base + v) == gn) ? 1.0f : 0.0f;
            xr[v] += omega * (idv - cr);
            xi[v] -= omega * ci;
        }

        // Publish new X for the next layer (skip after the final layer)
        if (l != nlayers - 1) {
            __syncthreads();   // everyone done reading old X
            #pragma unroll
            for (int v = 0; v < 8; ++v) {
                int xo = (gm_base + v) * LDS_STRIDE + gn;
                sXr[xo] = xr[v];
                sXi[xo] = xi[v];
            }
            __syncthreads();   // new X visible to all waves
        }
    }

    // ---- Write interleaved complex64 (re, im) straight from registers ----
    float* outB = out + (size_t)b * 4096 * 2;
    #pragma unroll
    for (int v = 0; v < 8; ++v) {
        int e = (gm_base + v) * 64 + gn;     // complex element index
        v2f vv; vv.x = xr[v]; vv.y = xi[v];
        *(v2f*)&outB[2 * e] = vv;
    }
}

extern "C" void kernel_launch(void* const* d_in, const int* in_sizes, int n_in,
                              void* d_out, int out_size, void* d_ws, size_t ws_size,
                              hipStream_t stream) {
    const float* Ar = (const float*)d_in[0];   // [B,64,64] f32
    const float* Ai = (const float*)d_in[1];   // [B,64,64] f32
    const float* om = (const float*)d_in[2];   // [nlayers] f32
    float* outp = (float*)d_out;               // [B,64,64] complex64 -> interleaved f32

    const int B       = in_sizes[0] / 4096;
    const int nlayers = in_sizes[2];
    const size_t shmem = (size_t)(4 * MAT_FLOATS) * sizeof(float); // ~74 KB

    hipFuncSetAttribute((const void*)richardson_dun_kernel,
                        hipFuncAttributeMaxDynamicSharedMemorySize, (int)shmem);

    richardson_dun_kernel<<<B, 512, shmem, stream>>>(Ar, Ai, om, nlayers, outp);
}